// GraphEmbedding_66202625901264
// MI455X (gfx1250) — compile-verified
//
#include <hip/hip_runtime.h>
#include <hip/hip_bf16.h>

#define D_DIM 64
#define SCAN_B 256   // scan block size; requires cdiv(N,256) <= 256 (N=50000 -> 196 blocks)
#define PCHUNK 16    // pooling partial chunks per graph

typedef float v2f __attribute__((ext_vector_type(2)));
typedef float v8f __attribute__((ext_vector_type(8)));

static inline int cdiv(long long a, long long b) { return (int)((a + b - 1) / b); }

// ---------------- CSR build: histogram of dst (native int atomics) ----------------
__global__ void k_hist(const int* __restrict__ dst, int* __restrict__ cnt, int E) {
  int e = blockIdx.x * blockDim.x + threadIdx.x;
  if (e < E) atomicAdd(&cnt[dst[e]], 1);
}

// ---------------- scan stage 1: per-block inclusive scan ----------------
__global__ void k_scan1(const int* __restrict__ cnt, int* __restrict__ incl,
                        int* __restrict__ bsum, int N) {
  __shared__ int sh[SCAN_B];
  int t = threadIdx.x;
  int i = blockIdx.x * SCAN_B + t;
  int v = (i < N) ? cnt[i] : 0;
  sh[t] = v;
  __syncthreads();
  for (int off = 1; off < SCAN_B; off <<= 1) {
    int u = (t >= off) ? sh[t - off] : 0;
    __syncthreads();
    sh[t] += u;
    __syncthreads();
  }
  if (i < N) incl[i] = sh[t];
  if (t == SCAN_B - 1) bsum[blockIdx.x] = sh[t];
}

// ---------------- scan stage 2: scan of block sums (single block) ----------------
__global__ void k_scan2(const int* __restrict__ bsum, int* __restrict__ boff, int nb) {
  __shared__ int sh[SCAN_B];
  int t = threadIdx.x;
  int v = (t < nb) ? bsum[t] : 0;
  sh[t] = v;
  __syncthreads();
  for (int off = 1; off < SCAN_B; off <<= 1) {
    int u = (t >= off) ? sh[t - off] : 0;
    __syncthreads();
    sh[t] += u;
    __syncthreads();
  }
  boff[t] = sh[t] - v;   // exclusive offset for block t
}

// ---------------- scan stage 3: row starts + fill cursors ----------------
__global__ void k_scan3(const int* __restrict__ cnt, const int* __restrict__ incl,
                        const int* __restrict__ boff, int* __restrict__ rowstart,
                        int* __restrict__ cur, int N, int E) {
  int i = blockIdx.x * blockDim.x + threadIdx.x;
  if (i < N) {
    int excl = incl[i] - cnt[i] + boff[i >> 8];   // >>8 == /SCAN_B
    rowstart[i] = excl;
    cur[i] = excl;
    if (i == 0) rowstart[N] = E;
  }
}

// ---------------- CSR fill: sorted-by-dst src ids + edge weights ----------------
__global__ void k_fill(const int* __restrict__ src, const int* __restrict__ dst,
                       const float* __restrict__ ea, int* __restrict__ cur,
                       int* __restrict__ ssrc, float* __restrict__ sea, int E) {
  int e = blockIdx.x * blockDim.x + threadIdx.x;
  if (e >= E) return;
  int p = atomicAdd(&cur[dst[e]], 1);
  ssrc[p] = src[e];
  sea[p] = ea[e];
}

// ---------------- per-node scalars from CSR (atomic-free) ----------------
__global__ void k_deg_csr(const int* __restrict__ rowstart, const float* __restrict__ sea,
                          float* __restrict__ dis, float* __restrict__ idg, int N) {
  int i = blockIdx.x * blockDim.x + threadIdx.x;
  if (i >= N) return;
  int beg = rowstart[i], end = rowstart[i + 1];
  float s = 0.0f;
  for (int j = beg; j < end; ++j) s += sea[j];
  float d = s + 1.0f;
  dis[i] = 1.0f / sqrtf(d);
  idg[i] = 1.0f / d;
}

// ---------------- fold dis[src] into sorted edge weights ----------------
__global__ void k_scale(const int* __restrict__ ssrc, const float* __restrict__ dis,
                        float* __restrict__ sea, int E) {
  int p = blockIdx.x * blockDim.x + threadIdx.x;
  if (p < E) sea[p] *= dis[ssrc[p]];
}

// ---------------- embedding gather: h[i] = emb[x[i]] ----------------
__global__ void k_gather(const int* __restrict__ x, const float* __restrict__ emb,
                         float* __restrict__ h, int N) {
  int t = blockIdx.x * blockDim.x + threadIdx.x;
  int node = t >> 4, c = t & 15;
  if (node < N) {
    const float4* s = (const float4*)(emb + (size_t)x[node] * D_DIM);
    ((float4*)(h + (size_t)node * D_DIM))[c] = s[c];
  }
}

// ---------------- hW = h @ W via V_WMMA_F32_16X16X4_F32 ----------------
// grid.x = N/16 (M tiles), blockDim = 128 (4 waves = 4 column tiles)
// A 16x4 f32: lane L: M=L&15, VGPR j holds K=2*(L>>4)+j
// B 4x16 f32: lane L: N=L&15, VGPR j holds K=2*(L>>4)+j
// C/D 16x16 f32: VGPR v: M=v+8*(L>>4), N=L&15
__global__ void k_gemm_wmma(const float* __restrict__ h, const float* __restrict__ W,
                            float* __restrict__ out) {
  const int lane = threadIdx.x & 31;
  const int ncol = (threadIdx.x >> 5) << 4;
  const int mrow = blockIdx.x << 4;
  const int mlo  = lane & 15;
  const int khi  = (lane >> 4) << 1;               // 0 or 2
  const float* arow = h + (size_t)(mrow + mlo) * D_DIM + khi;
  const float* bcol = W + (size_t)khi * D_DIM + ncol + mlo;
  v8f acc = {};
#pragma unroll
  for (int k0 = 0; k0 < D_DIM; k0 += 4) {
    v2f a = *(const v2f*)(arow + k0);
    v2f b;
    b.x = bcol[(size_t)k0 * D_DIM];
    b.y = bcol[(size_t)k0 * D_DIM + D_DIM];
    acc = __builtin_amdgcn_wmma_f32_16x16x4_f32(false, a, false, b,
                                                (short)0, acc, false, false);
  }
  float* orow = out + (size_t)(mrow + (khi << 2)) * D_DIM + ncol + mlo;
#pragma unroll
  for (int v = 0; v < 8; ++v) orow[(size_t)v * D_DIM] = acc[v];
}

// ---------------- fused aggregate + self-loop + bias (+relu): atomic-free ----------------
// one wave per node; float2 per lane covers the 64-wide row
__global__ void k_aggregate(const int* __restrict__ rowstart, const int* __restrict__ ssrc,
                            const float* __restrict__ sea, const float* __restrict__ dis,
                            const float* __restrict__ idg, const float* __restrict__ hW,
                            const float* __restrict__ bias, float* __restrict__ h,
                            int relu, int N) {
  int t = blockIdx.x * blockDim.x + threadIdx.x;
  int node = t >> 5;
  if (node >= N) return;
  int lane = t & 31;
  int beg = rowstart[node], end = rowstart[node + 1];
  float dn  = dis[node];
  float idn = idg[node];
  float ax = 0.0f, ay = 0.0f;
  for (int j = beg; j < end; ++j) {
    int s = ssrc[j];
    float w = sea[j] * dn;                          // = dis[src]*ea*dis[dst]
    v2f v = *(const v2f*)(hW + (size_t)s * D_DIM + (lane << 1));
    ax += v.x * w;
    ay += v.y * w;
  }
  v2f self = *(const v2f*)(hW + (size_t)node * D_DIM + (lane << 1));
  v2f bb   = *(const v2f*)(bias + (lane << 1));
  float rx = ax + self.x * idn + bb.x;
  float ry = ay + self.y * idn + bb.y;
  if (relu) { rx = fmaxf(rx, 0.0f); ry = fmaxf(ry, 0.0f); }
  v2f r; r.x = rx; r.y = ry;
  *(v2f*)(h + (size_t)node * D_DIM + (lane << 1)) = r;
}

// ---------------- graph boundaries: batch is sorted; binary search ----------------
__global__ void k_bounds(const int* __restrict__ batch, int* __restrict__ bound,
                         int N, int G) {
  int g = blockIdx.x * blockDim.x + threadIdx.x;
  if (g > G) return;
  int lo = 0, hi = N;                  // lower_bound of g
  while (lo < hi) {
    int mid = (lo + hi) >> 1;
    if (batch[mid] < g) lo = mid + 1; else hi = mid;
  }
  bound[g] = lo;
}

// ---------------- pooling partials: one wave per (graph, chunk), no atomics ----------------
__global__ void k_pool_seg(const float* __restrict__ h, const int* __restrict__ bound,
                           float* __restrict__ part, int G) {
  int t = blockIdx.x * blockDim.x + threadIdx.x;
  int w = t >> 5;
  if (w >= G * PCHUNK) return;
  int lane = t & 31;
  int g = w / PCHUNK, c = w - g * PCHUNK;
  int beg = bound[g], end = bound[g + 1];
  long long len = end - beg;
  int s = beg + (int)((len * c) / PCHUNK);
  int e = beg + (int)((len * (c + 1)) / PCHUNK);
  float ax = 0.0f, ay = 0.0f;
  for (int i = s; i < e; ++i) {
    v2f v = *(const v2f*)(h + (size_t)i * D_DIM + (lane << 1));
    ax += v.x;
    ay += v.y;
  }
  v2f r; r.x = ax; r.y = ay;
  *(v2f*)(part + (size_t)w * D_DIM + (lane << 1)) = r;
}

// ---------------- finalize: reduce partials, mean, L2 normalize. grid=G, block=64 ----------------
__global__ void k_finalize(const float* __restrict__ part, const int* __restrict__ bound,
                           float* __restrict__ out) {
  int g = blockIdx.x;
  int d = threadIdx.x;
  float s = 0.0f;
#pragma unroll
  for (int c = 0; c < PCHUNK; ++c)
    s += part[((size_t)g * PCHUNK + c) * D_DIM + d];
  float cntf = (float)(bound[g + 1] - bound[g]);
  float v = s / fmaxf(cntf, 1.0f);
  __shared__ float ss[D_DIM];
  ss[d] = v * v;
  __syncthreads();
  for (int off = 32; off > 0; off >>= 1) {
    if (d < off) ss[d] += ss[d + off];
    __syncthreads();
  }
  float nrm = sqrtf(ss[0]);
  out[(size_t)g * D_DIM + d] = v / fmaxf(nrm, 1e-12f);
}

extern "C" void kernel_launch(void* const* d_in, const int* in_sizes, int n_in,
                              void* d_out, int out_size, void* d_ws, size_t ws_size,
                              hipStream_t stream) {
  const int*   x    = (const int*)d_in[0];
  const int*   ei   = (const int*)d_in[1];
  const float* ea   = (const float*)d_in[2];
  const int*   bat  = (const int*)d_in[3];
  const float* emb  = (const float*)d_in[4];
  const float* Wm[3] = {(const float*)d_in[5], (const float*)d_in[7], (const float*)d_in[9]};
  const float* bv[3] = {(const float*)d_in[6], (const float*)d_in[8], (const float*)d_in[10]};

  const int N = in_sizes[0];
  const int E = in_sizes[2];
  const int G = out_size / D_DIM;
  const int* src = ei;
  const int* dst = ei + E;
  const int nb = cdiv(N, SCAN_B);   // must be <= SCAN_B (N=50000 -> 196)

  // workspace carve-out (byte-based, 256B-aligned slots)
  size_t off = 0;
  char* base = (char*)d_ws;
  auto carve = [&](size_t bytes) -> void* {
    void* r = base + off;
    off += (bytes + 255) & ~(size_t)255;
    return r;
  };
  int*   cnt      = (int*)  carve((size_t)N * 4);
  int*   incl     = (int*)  carve((size_t)N * 4);
  int*   bsum     = (int*)  carve((size_t)SCAN_B * 4);
  int*   boff     = (int*)  carve((size_t)SCAN_B * 4);
  int*   rowstart = (int*)  carve((size_t)(N + 1) * 4);
  int*   cur      = (int*)  carve((size_t)N * 4);
  int*   ssrc     = (int*)  carve((size_t)E * 4);
  float* sea      = (float*)carve((size_t)E * 4);
  float* dis      = (float*)carve((size_t)N * 4);
  float* idg      = (float*)carve((size_t)N * 4);
  float* h        = (float*)carve((size_t)N * D_DIM * 4);
  float* hW       = (float*)carve((size_t)N * D_DIM * 4);
  int*   bound    = (int*)  carve((size_t)(G + 1) * 4);
  float* part     = (float*)carve((size_t)G * PCHUNK * D_DIM * 4);
  (void)ws_size; (void)n_in;

  // ---- build CSR by dst (reused by all 3 layers) ----
  hipMemsetAsync(cnt, 0, (size_t)N * 4, stream);
  k_hist <<<cdiv(E, 256), 256, 0, stream>>>(dst, cnt, E);
  k_scan1<<<nb, SCAN_B, 0, stream>>>(cnt, incl, bsum, N);
  k_scan2<<<1, SCAN_B, 0, stream>>>(bsum, boff, nb);
  k_scan3<<<cdiv(N, 256), 256, 0, stream>>>(cnt, incl, boff, rowstart, cur, N, E);
  k_fill <<<cdiv(E, 256), 256, 0, stream>>>(src, dst, ea, cur, ssrc, sea, E);

  // ---- node scalars + normalized edge weights + input embeddings ----
  k_deg_csr<<<cdiv(N, 256), 256, 0, stream>>>(rowstart, sea, dis, idg, N);
  k_scale  <<<cdiv(E, 256), 256, 0, stream>>>(ssrc, dis, sea, E);
  k_gather <<<cdiv((long long)N * 16, 256), 256, 0, stream>>>(x, emb, h, N);

  // ---- 3 GCN layers: WMMA GEMM + atomic-free fused aggregation ----
  for (int l = 0; l < 3; ++l) {
    k_gemm_wmma<<<N / 16, 128, 0, stream>>>(h, Wm[l], hW);   // N % 16 == 0 (50000)
    k_aggregate<<<cdiv((long long)N * 32, 256), 256, 0, stream>>>(
        rowstart, ssrc, sea, dis, idg, hW, bv[l], h, (l < 2) ? 1 : 0, N);
  }

  // ---- mean pool + L2 normalize (atomic-free: batch is sorted) ----
  k_bounds  <<<cdiv(G + 1, 128), 128, 0, stream>>>(bat, bound, N, G);
  k_pool_seg<<<cdiv((long long)G * PCHUNK * 32, 256), 256, 0, stream>>>(h, bound, part, G);
  k_finalize<<<G, D_DIM, 0, stream>>>(part, bound, (float*)d_out);
}